// RotationLoss_26250840113463
// MI455X (gfx1250) — compile-verified
//
#include <hip/hip_runtime.h>
#include <math.h>

// ---------------------------------------------------------------------------
// RotationLoss on MI455X (gfx1250, wave32)
//
// Math: Procrustes-to-SO(3) with det-sign fix == Davenport q-method:
//   q* = dominant eigenvector of symmetric 4x4 K(M).
//   tr(Rp^T Rt) = 4*(qp.qt)^2 - 1  =>  chordal = 8(1-d^2), cos = 2d^2-1.
// Eigenvector via shifted repeated squaring of (K + sqrt(3)*||M||_F * I).
//
// Data path: TDM tensor_load_to_lds stages each block's 2x9KB tile into LDS
// (s_wait_tensorcnt), lanes read 9 floats at dword-stride 9 (conflict-free).
// Reduction: v_wmma_f32_16x16x4_f32 with all-ones B = wave32 row-sum reduce.
// ---------------------------------------------------------------------------

#define TILE 256
#define NSQ  6

typedef __attribute__((ext_vector_type(4))) unsigned int u32x4;
typedef __attribute__((ext_vector_type(8))) int          i32x8;
typedef __attribute__((ext_vector_type(4))) int          i32x4;
typedef __attribute__((ext_vector_type(2))) float        v2f;
typedef __attribute__((ext_vector_type(8))) float        v8f;

// ---------------- TDM issue: 1-row tensor, tile == tensor, 4B elements ------
#if __has_builtin(__builtin_amdgcn_tensor_load_to_lds)
__device__ __forceinline__ void tdm_issue(unsigned lds_off, const float* gsrc,
                                          unsigned nfloats) {
  unsigned long long ga = (unsigned long long)(size_t)gsrc;
  u32x4 g0;
  g0[0] = 1u;                                        // count=1, user mode
  g0[1] = lds_off;                                   // lds_addr (bytes)
  g0[2] = (unsigned)(ga & 0xffffffffu);              // global_addr[31:0]
  g0[3] = (unsigned)((ga >> 32) & 0x01ffffffu)       // global_addr[56:32]
        | (2u << 30);                                // type=2 ("image")
  i32x8 g1;
  g1[0] = (int)(2u << 16);                           // data_size=2 -> 4 bytes
  g1[1] = (int)((nfloats & 0xffffu) << 16);          // tensor_dim0 lo16
  g1[2] = (int)(((nfloats >> 16) & 0xffffu)          // tensor_dim0 hi16
        | (1u << 16));                               // tensor_dim1 = 1 (lo16)
  g1[3] = (int)((nfloats & 0xffffu) << 16);          // tile_dim0 = nfloats
  g1[4] = 1;                                         // tile_dim1=1, tile_dim2=0
  g1[5] = (int)nfloats;                              // tensor_dim0_stride lo32
  g1[6] = 0;
  g1[7] = 0;
  i32x4 z4 = {0, 0, 0, 0};                           // <=2D: groups 2/3 unused
  i32x8 z8 = {0, 0, 0, 0, 0, 0, 0, 0};               // extra group (6-arg form)
  __builtin_amdgcn_tensor_load_to_lds(g0, g1, z4, z4, z8, 0);
}
#endif

// ---------------- dominant eigenvector of Davenport K --------------------
// Returns unnormalized quaternion; caller uses dot/norm.
__device__ __forceinline__ float4 best_quat(const float* __restrict__ m) {
  const float Sxx = m[0], Sxy = m[1], Sxz = m[2];
  const float Syx = m[3], Syy = m[4], Syz = m[5];
  const float Szx = m[6], Szy = m[7], Szz = m[8];

  const float fn2 = Sxx*Sxx + Sxy*Sxy + Sxz*Sxz + Syx*Syx + Syy*Syy
                  + Syz*Syz + Szx*Szx + Szy*Szy + Szz*Szz;
  const float c = 1.7320508f * sqrtf(fn2) + 1e-20f;  // shift: K+cI is PSD

  float p00 =  Sxx + Syy + Szz + c;
  float p01 =  Syz - Szy;
  float p02 =  Szx - Sxz;
  float p03 =  Sxy - Syx;
  float p11 =  Sxx - Syy - Szz + c;
  float p12 =  Sxy + Syx;
  float p13 =  Szx + Sxz;
  float p22 = -Sxx + Syy - Szz + c;
  float p23 =  Syz + Szy;
  float p33 = -Sxx - Syy + Szz + c;

#pragma unroll
  for (int it = 0; it < NSQ; ++it) {                 // P <- P^2 / tr(P^2)
    const float n00 = p00*p00 + p01*p01 + p02*p02 + p03*p03;
    const float n01 = p00*p01 + p01*p11 + p02*p12 + p03*p13;
    const float n02 = p00*p02 + p01*p12 + p02*p22 + p03*p23;
    const float n03 = p00*p03 + p01*p13 + p02*p23 + p03*p33;
    const float n11 = p01*p01 + p11*p11 + p12*p12 + p13*p13;
    const float n12 = p01*p02 + p11*p12 + p12*p22 + p13*p23;
    const float n13 = p01*p03 + p11*p13 + p12*p23 + p13*p33;
    const float n22 = p02*p02 + p12*p12 + p22*p22 + p23*p23;
    const float n23 = p02*p03 + p12*p13 + p22*p23 + p23*p33;
    const float n33 = p03*p03 + p13*p13 + p23*p23 + p33*p33;
    const float inv = 1.0f / (n00 + n11 + n22 + n33);
    p00 = n00*inv; p01 = n01*inv; p02 = n02*inv; p03 = n03*inv;
    p11 = n11*inv; p12 = n12*inv; p13 = n13*inv;
    p22 = n22*inv; p23 = n23*inv;
    p33 = n33*inv;
  }
  // P ~ q q^T : take the column with the largest diagonal (branchless cndmask)
  float best = p00, q0 = p00, q1 = p01, q2 = p02, q3 = p03;
  if (p11 > best) { best = p11; q0 = p01; q1 = p11; q2 = p12; q3 = p13; }
  if (p22 > best) { best = p22; q0 = p02; q1 = p12; q2 = p22; q3 = p23; }
  if (p33 > best) {             q0 = p03; q1 = p13; q2 = p23; q3 = p33; }
  return make_float4(q0, q1, q2, q3);
}

// ---------------- wave32 sum reduction via WMMA (ones-B row-sum trick) ----
__device__ __forceinline__ float wave_reduce_sum(float v) {
#if __has_builtin(__builtin_amdgcn_wmma_f32_16x16x4_f32)
  v2f a; a[0] = v;    a[1] = 0.0f;      // A[16x4]: col0/col2 hold the 32 lanes
  v2f b; b[0] = 1.0f; b[1] = 1.0f;      // B[4x16] = all ones
  v8f cacc = {0.f,0.f,0.f,0.f,0.f,0.f,0.f,0.f};
  v8f d = __builtin_amdgcn_wmma_f32_16x16x4_f32(false, a, false, b,
                                                (short)0, cacc, false, false);
  // lanes 0-15 hold rowsums 0..7, lanes 16-31 hold rowsums 8..15
  float s = d[0] + d[1] + d[2] + d[3] + d[4] + d[5] + d[6] + d[7];
  s += __shfl_xor(s, 16, 32);
  return s;
#else
  for (int off = 16; off > 0; off >>= 1) v += __shfl_xor(v, off, 32);
  return v;
#endif
}

// ---------------- kernel 1: per-block partial loss ------------------------
__global__ __launch_bounds__(TILE) void rotloss_partial_kernel(
    const float* __restrict__ pred, const float* __restrict__ target,
    float* __restrict__ partial, int B) {
  __shared__ float sP[TILE * 9];
  __shared__ float sT[TILE * 9];
  __shared__ float sWave[TILE / 32];

  const int base = blockIdx.x * TILE;
  const int nf   = min(TILE * 9, (B - base) * 9);

#if __has_builtin(__builtin_amdgcn_tensor_load_to_lds)
  if (threadIdx.x == 0) {
    tdm_issue((unsigned)(size_t)(void*)sP, pred   + (size_t)base * 9, (unsigned)nf);
    tdm_issue((unsigned)(size_t)(void*)sT, target + (size_t)base * 9, (unsigned)nf);
    __builtin_amdgcn_s_wait_tensorcnt((short)0);     // TDM -> LDS complete
  }
#else
  for (int i = threadIdx.x; i < nf; i += TILE) {
    sP[i] = pred  [(size_t)base * 9 + i];
    sT[i] = target[(size_t)base * 9 + i];
  }
#endif
  __syncthreads();

  // dword-stride-9 LDS reads: gcd(9,64)=1 -> bank-conflict-free across wave32
  const float4 qp = best_quat(&sP[threadIdx.x * 9]);
  const float4 qt = best_quat(&sT[threadIdx.x * 9]);
  const float np  = qp.x*qp.x + qp.y*qp.y + qp.z*qp.z + qp.w*qp.w;
  const float nt  = qt.x*qt.x + qt.y*qt.y + qt.z*qt.z + qt.w*qt.w;
  const float dq  = qp.x*qt.x + qp.y*qt.y + qp.z*qt.z + qp.w*qt.w;
  const float dn  = dq * rsqrtf(np * nt + 1e-30f);
  const float d2  = dn * dn;
  float cosT = 2.0f * d2 - 1.0f;                               // (tr-1)/2
  cosT = fminf(fmaxf(cosT, -1.0f + 1e-7f), 1.0f - 1e-7f);
  const float ang = acosf(cosT) * 57.295779513082323f;         // degrees
  float L = 8.0f * (1.0f - d2) + 0.1f * ang;                   // chordal+0.1*ang
  L = ((base + (int)threadIdx.x) < B) ? L : 0.0f;              // tail guard

  // full-EXEC wave reduce (WMMA), then 8 wave partials -> 1 block partial
  const float wsum = wave_reduce_sum(L);
  const int lane = threadIdx.x & 31;
  const int wid  = threadIdx.x >> 5;
  if (lane == 0) sWave[wid] = wsum;
  __syncthreads();
  if (threadIdx.x == 0) {
    float s = 0.0f;
#pragma unroll
    for (int w = 0; w < TILE / 32; ++w) s += sWave[w];
    partial[blockIdx.x] = s;
  }
}

// ---------------- kernel 2: deterministic final reduction -----------------
__global__ __launch_bounds__(256) void rotloss_final_kernel(
    const float* __restrict__ partial, int nparts, float invB,
    float* __restrict__ out) {
  __shared__ float sr[256];
  float s = 0.0f;
  for (int i = threadIdx.x; i < nparts; i += 256) s += partial[i];
  sr[threadIdx.x] = s;
  __syncthreads();
  for (int stride = 128; stride > 0; stride >>= 1) {
    if ((int)threadIdx.x < stride) sr[threadIdx.x] += sr[threadIdx.x + stride];
    __syncthreads();
  }
  if (threadIdx.x == 0) out[0] = sr[0] * invB;
}

// ---------------- launch ---------------------------------------------------
extern "C" void kernel_launch(void* const* d_in, const int* in_sizes, int n_in,
                              void* d_out, int out_size, void* d_ws, size_t ws_size,
                              hipStream_t stream) {
  const float* pred   = (const float*)d_in[0];
  const float* target = (const float*)d_in[1];
  const int B = in_sizes[0] / 9;
  const int nblocks = (B + TILE - 1) / TILE;
  float* partial = (float*)d_ws;

  rotloss_partial_kernel<<<nblocks, TILE, 0, stream>>>(pred, target, partial, B);
  rotloss_final_kernel<<<1, 256, 0, stream>>>(partial, nblocks,
                                              1.0f / (float)B, (float*)d_out);
}